// FusedExpertMixer_6150393168450
// MI455X (gfx1250) — compile-verified
//
#include <hip/hip_runtime.h>
#include <hip/hip_bf16.h>

// ---------------------------------------------------------------------------
// FusedExpertMixer for MI455X (gfx1250)
//   out[t, :] = w0[t] * stack[e0[t], t, :] + w1[t] * stack[e1[t], t, :]
// Memory-bound (0.25 FLOP/byte, ~192 MiB streamed -> ~8.6 us floor @23.3TB/s):
//   - 128-bit coalesced accesses (256 thr x float4 == H)
//   - CDNA5 async global->LDS loads (ASYNCcnt) for the gathered rows
//   - non-temporal hints (zero reuse; don't thrash the 192MB L2)
// ---------------------------------------------------------------------------

typedef float v4f __attribute__((ext_vector_type(4)));
typedef int   v4i __attribute__((vector_size(16)));   // matches builtin param type

#define GAS __attribute__((address_space(1)))
#define LAS __attribute__((address_space(3)))

#if __has_builtin(__builtin_amdgcn_global_load_async_to_lds_b128) && \
    __has_builtin(__builtin_amdgcn_s_wait_asynccnt)
#define USE_ASYNC_LDS 1
#else
#define USE_ASYNC_LDS 0
#endif

namespace {
constexpr int kE = 8;
constexpr int kB = 4;
constexpr int kS = 4096;
constexpr int kH = 1024;
constexpr int kK = 2;
constexpr int kTok = kB * kS;      // 16384 tokens
constexpr int kThreads = 256;      // 8 wave32s; 256*4 floats == kH
}  // namespace

__global__ __launch_bounds__(kThreads) void moe_mix_kernel(
    const float* __restrict__ stack,   // [E, B*S, H]
    const float* __restrict__ wgt,     // [B*S, K]
    const int*   __restrict__ idx,     // [B*S, K]
    float* __restrict__ out)           // [B*S, H]
{
    const int t   = blockIdx.x;        // token id in [0, kTok)
    const int tid = threadIdx.x;       // [0, 256)
    const int h   = tid * 4;           // float4 slot within the row

    // Block-uniform routing data -> scalar loads + wave broadcast.
    const int   e0 = idx[(size_t)t * kK + 0];
    const int   e1 = idx[(size_t)t * kK + 1];
    const float w0 = wgt[(size_t)t * kK + 0];
    const float w1 = wgt[(size_t)t * kK + 1];

    const size_t rowA = ((size_t)e0 * kTok + t) * kH;
    const size_t rowB = ((size_t)e1 * kTok + t) * kH;

    v4f a, b;

#if USE_ASYNC_LDS
    // Stage both gathered rows through LDS with the CDNA5 async copy engine.
    // Each lane loads exactly the 16B it will read back, so a per-wave
    // s_wait_asynccnt(0) is sufficient (no cross-wave sharing -> no barrier).
    __shared__ float lbuf[2 * kH];     // 8 KB staging
    __builtin_amdgcn_global_load_async_to_lds_b128(
        (GAS v4i*)(stack + rowA + h),
        (LAS v4i*)(&lbuf[h]),
        /*offset=*/0, /*cpol=*/0);
    __builtin_amdgcn_global_load_async_to_lds_b128(
        (GAS v4i*)(stack + rowB + h),
        (LAS v4i*)(&lbuf[kH + h]),
        /*offset=*/0, /*cpol=*/0);
    __builtin_amdgcn_s_wait_asynccnt(0);
    a = *(const v4f*)(&lbuf[h]);
    b = *(const v4f*)(&lbuf[kH + h]);
#else
    // Fallback: direct non-temporal 128-bit gathers to VGPRs.
    a = __builtin_nontemporal_load((const v4f*)(stack + rowA + h));
    b = __builtin_nontemporal_load((const v4f*)(stack + rowB + h));
#endif

    v4f r;
    r.x = fmaf(w0, a.x, w1 * b.x);
    r.y = fmaf(w0, a.y, w1 * b.y);
    r.z = fmaf(w0, a.z, w1 * b.z);
    r.w = fmaf(w0, a.w, w1 * b.w);

    // Streaming store, no reuse -> non-temporal b128.
    __builtin_nontemporal_store(r, (v4f*)(out + (size_t)t * kH + h));
}

extern "C" void kernel_launch(void* const* d_in, const int* in_sizes, int n_in,
                              void* d_out, int out_size, void* d_ws, size_t ws_size,
                              hipStream_t stream) {
    (void)in_sizes; (void)n_in; (void)d_ws; (void)ws_size; (void)out_size;
    // setup_inputs order: [0]=hidden_states (UNUSED by reference),
    // [1]=expert_stack [E,B,S,H], [2]=expert_weights [B,S,K], [3]=expert_indices [B,S,K]
    const float* stack = (const float*)d_in[1];
    const float* wgt   = (const float*)d_in[2];
    const int*   idx   = (const int*)d_in[3];
    float*       out   = (float*)d_out;

    moe_mix_kernel<<<dim3(kTok), dim3(kThreads), 0, stream>>>(stack, wgt, idx, out);
}